// FinalNMSLoss_87136296501791
// MI455X (gfx1250) — compile-verified
//
#include <hip/hip_runtime.h>
#include <stdint.h>

// ---------------- problem constants (match reference) ----------------
static constexpr int   BB   = 2;
static constexpr int   NN   = 4096;
static constexpr int   GG   = 64;
static constexpr int   TT   = 1024;          // threads per block (32 waves, wave32)
static constexpr int   PER  = NN / TT;       // 4 proposals per thread
static constexpr float NMS_THR = 0.5f;
static constexpr float EPSV    = 1e-6f;

// ---------------- LDS layout (dynamic shared, ~163 KB; CDNA5 WGP has 320 KB) --
static constexpr uint32_t OFF_BOXES  = 0;                                // 4*N f32
static constexpr uint32_t OFF_SCORE  = OFF_BOXES  + 4u*NN*4u;            // N f32
static constexpr uint32_t OFF_AREA   = OFF_SCORE  + NN*4u;               // N f32
static constexpr uint32_t OFF_AGT    = OFF_AREA   + NN*4u;               // N i32
static constexpr uint32_t OFF_GTB    = OFF_AGT    + NN*4u;               // G*4 f32
static constexpr uint32_t OFF_GTAREA = OFF_GTB    + GG*4u*4u;            // G f32
static constexpr uint32_t OFF_GTIOU  = OFF_GTAREA + GG*4u;               // G*G f32
static constexpr uint32_t OFF_SKEY   = OFF_GTIOU  + GG*GG*4u;            // N f32
static constexpr uint32_t OFF_ORDER  = OFF_SKEY   + NN*4u;               // N i32
static constexpr uint32_t OFF_ACT    = OFF_ORDER  + NN*4u;               // N/32 u32
static constexpr uint32_t OFF_MREC   = OFF_ACT    + (NN/32u)*4u;         // G i32
static constexpr uint32_t OFF_REDS   = OFF_MREC   + GG*4u;               // 32 f32
static constexpr uint32_t OFF_REDC   = OFF_REDS   + 32u*4u;              // 32 u32
static constexpr uint32_t OFF_REDR   = OFF_REDC   + 32u*4u;              // 32 u32
static constexpr uint32_t OFF_MISC   = OFF_REDR   + 32u*4u;
static constexpr uint32_t SMEM_BYTES = OFF_MISC + 64u;

struct Misc {
    float tpull, tpush, pc, pushc;
    int   activeCount, npos0;
};

// ---------------- CDNA5 Tensor Data Mover path ----------------
#if defined(__HIP_DEVICE_COMPILE__) && __has_builtin(__builtin_amdgcn_tensor_load_to_lds)
#define USE_TDM 1
#else
#define USE_TDM 0
#endif

#if USE_TDM
typedef unsigned int v4u __attribute__((ext_vector_type(4)));
typedef int          v8i __attribute__((ext_vector_type(8)));
typedef int          v4i __attribute__((ext_vector_type(4)));

// 2D strided DMA: copies `rows` rows of `elems_per_row` f32 (row stride
// `row_stride_elems` in memory) into contiguous LDS at lds_addr.
// D# per CDNA5 ISA ch.8: group0 {count,lds_addr,global_addr,type=2},
// group1 {data_size=4B, tensor_dim0/1, tile_dim0/1, tensor_dim0_stride}.
__device__ __forceinline__ void tdm_load_2d(uint32_t lds_addr, const void* gptr,
                                            uint32_t elems_per_row, uint32_t rows,
                                            uint32_t row_stride_elems) {
    uint64_t ga = (uint64_t)(uintptr_t)gptr;
    v4u g0;
    g0.x = 1u;                                                  // count=1, user mode
    g0.y = lds_addr;                                            // LDS byte address
    g0.z = (uint32_t)(ga & 0xFFFFFFFFu);                        // global_addr[31:0]
    g0.w = (uint32_t)((ga >> 32) & 0x1FFFFFFu) | (2u << 30);    // addr[56:32] | type=2
    v8i g1;
    g1[0] = (int)(2u << 16);                                    // data_size=2 (4 bytes)
    g1[1] = (int)((elems_per_row & 0xFFFFu) << 16);             // tensor_dim0[15:0]
    g1[2] = (int)(((elems_per_row >> 16) & 0xFFFFu) |
                  ((rows & 0xFFFFu) << 16));                    // dim0 hi | tensor_dim1 lo
    g1[3] = (int)(((rows >> 16) & 0xFFFFu) |
                  ((elems_per_row & 0xFFFFu) << 16));           // dim1 hi | tile_dim0
    g1[4] = (int)(rows & 0xFFFFu);                              // tile_dim1 (tile_dim2=0)
    g1[5] = (int)row_stride_elems;                              // tensor_dim0_stride[31:0]
    g1[6] = 0;
    g1[7] = 0;
    v4i z4 = {0, 0, 0, 0};
#if __clang_major__ >= 23
    v8i z8 = {0, 0, 0, 0, 0, 0, 0, 0};
    __builtin_amdgcn_tensor_load_to_lds(g0, g1, z4, z4, z8, 0);
#else
    __builtin_amdgcn_tensor_load_to_lds(g0, g1, z4, z4, 0);
#endif
}
#endif // USE_TDM

// IoU with the reference's "+1" (Pascal) convention.
__device__ __forceinline__ float iou_box(const float* __restrict__ bx,
                                         const float* __restrict__ ar, int a, int b) {
    const float* A = bx + 4 * a;
    const float* Bp = bx + 4 * b;
    float lx = fmaxf(A[0], Bp[0]);
    float ly = fmaxf(A[1], Bp[1]);
    float rx = fminf(A[2], Bp[2]);
    float ry = fminf(A[3], Bp[3]);
    float w  = fmaxf(rx - lx + 1.0f, 0.0f);
    float h  = fmaxf(ry - ly + 1.0f, 0.0f);
    float ov = w * h;
    return ov / (ar[a] + ar[b] - ov);
}

__device__ __forceinline__ bool sort_before(float sa, int ia, float sb, int ib) {
    // descending score, stable (ascending original index on ties)
    return (sa > sb) || (sa == sb && ia < ib);
}

__global__ void __launch_bounds__(TT, 1)
nms_loss_kernel(const int* __restrict__ agt_g, const float* __restrict__ gtb_g,
                const float* __restrict__ prop_g, float* __restrict__ ws) {
    extern __shared__ char smem[];
    float*    bxs  = (float*)(smem + OFF_BOXES);
    float*    scr  = (float*)(smem + OFF_SCORE);
    float*    area = (float*)(smem + OFF_AREA);
    int*      agt  = (int*)(smem + OFF_AGT);
    float*    gbx  = (float*)(smem + OFF_GTB);
    float*    gar  = (float*)(smem + OFF_GTAREA);
    float*    giou = (float*)(smem + OFF_GTIOU);
    float*    skey = (float*)(smem + OFF_SKEY);
    int*      ord  = (int*)(smem + OFF_ORDER);
    unsigned* act  = (unsigned*)(smem + OFF_ACT);
    int*      mrec = (int*)(smem + OFF_MREC);
    float*    redS = (float*)(smem + OFF_REDS);
    unsigned* redC = (unsigned*)(smem + OFF_REDC);
    unsigned* redR = (unsigned*)(smem + OFF_REDR);
    Misc*     M    = (Misc*)(smem + OFF_MISC);

    const int tid = threadIdx.x;
    const int b   = blockIdx.x;
    const int wid = tid >> 5;  // wave id (wave32)

    const int*   gAgt  = agt_g  + (size_t)b * NN;
    const float* gGt   = gtb_g  + (size_t)b * GG * 4;
    const float* gProp = prop_g + (size_t)b * NN * 5;

    // ---------------- phase 0: fill LDS ----------------
#if USE_TDM
    if (tid == 0) {
        // async DMA de-interleave: boxes (4 of every 5 floats) and scores (1 of 5)
        tdm_load_2d((uint32_t)(uintptr_t)(smem + OFF_BOXES), gProp, 4u, (uint32_t)NN, 5u);
        tdm_load_2d((uint32_t)(uintptr_t)(smem + OFF_SCORE), gProp + 4, 1u, (uint32_t)NN, 5u);
        __builtin_amdgcn_s_wait_tensorcnt(0);
    }
#else
    for (int t = 0; t < PER; ++t) {
        int j = tid + t * TT;
        const float* p = gProp + 5 * j;
        bxs[4 * j + 0] = p[0];
        bxs[4 * j + 1] = p[1];
        bxs[4 * j + 2] = p[2];
        bxs[4 * j + 3] = p[3];
        scr[j] = p[4];
    }
#endif
    if (tid == 1) {  // zero accumulators (thread 0's wave is busy issuing TDM)
        M->tpull = 0.0f; M->tpush = 0.0f; M->pc = 0.0f; M->pushc = 0.0f;
        M->activeCount = 0; M->npos0 = 0;
    }
    for (int t = 0; t < PER; ++t) {
        int j = tid + t * TT;
        agt[j] = gAgt[j];
    }
    if (tid < GG * 4) gbx[tid] = gGt[tid];
    __syncthreads();

    // ---------------- phase 1: areas, sort keys, init ----------------
    for (int t = 0; t < PER; ++t) {
        int j = tid + t * TT;
        float x1 = bxs[4 * j + 0], y1 = bxs[4 * j + 1];
        float x2 = bxs[4 * j + 2], y2 = bxs[4 * j + 3];
        area[j] = (x2 - x1 + 1.0f) * (y2 - y1 + 1.0f);
        skey[j] = scr[j];
        ord[j]  = j;
    }
    if (tid < GG) {
        gar[tid] = (gbx[4 * tid + 2] - gbx[4 * tid + 0] + 1.0f) *
                   (gbx[4 * tid + 3] - gbx[4 * tid + 1] + 1.0f);
        mrec[tid] = -1;
    }
    if (tid < NN / 32) {
        unsigned w = 0;
        int c = 0;
        for (int k = 0; k < 32; ++k) {
            if (agt[tid * 32 + k] >= 0) { w |= (1u << k); ++c; }
        }
        act[tid] = w;
        atomicAdd(&M->activeCount, c);
    }
    __syncthreads();

    // ---------------- phase 2: gt_iou (64x64) ----------------
    for (int t = 0; t < PER; ++t) {
        int p = tid + t * TT;  // 4096 pairs
        int r = p >> 6, c = p & 63;
        giou[p] = iou_box(gbx, gar, r, c);
    }
    if (tid == 0) M->npos0 = M->activeCount;

    // ---------------- phase 3: bitonic sort (stable argsort of -scores) ------
    for (unsigned k = 2; k <= (unsigned)NN; k <<= 1) {
        for (unsigned jj = k >> 1; jj > 0; jj >>= 1) {
            __syncthreads();
            for (int t = 0; t < PER; ++t) {
                unsigned i0 = (unsigned)(tid + t * TT);
                unsigned ix = i0 ^ jj;
                if (ix > i0) {
                    bool dirUp = ((i0 & k) == 0);
                    float sa = skey[i0]; int ia = ord[i0];
                    float sb = skey[ix]; int ib = ord[ix];
                    if (sort_before(sa, ia, sb, ib) != dirUp) {
                        skey[i0] = sb; ord[i0] = ib;
                        skey[ix] = sa; ord[ix] = ia;
                    }
                }
            }
        }
    }

    // ---------------- phase 4: sequential greedy scan ----------------
    for (int i = 0; i < NN; ++i) {
        __syncthreads();  // makes previous step's thread-0 updates visible
        const int  idx  = ord[i];
        const int  graw = agt[idx];
        const bool ai   = (act[idx >> 5] >> (idx & 31)) & 1u;  // uniform

        if (ai) {
            const int g = graw;  // ai implies graw >= 0
            // row pass: IoU(idx, j) over still-active j
            const float Ax1 = bxs[4 * idx + 0], Ay1 = bxs[4 * idx + 1];
            const float Ax2 = bxs[4 * idx + 2], Ay2 = bxs[4 * idx + 3];
            const float Aar = area[idx];
            float    ls = 0.0f;
            unsigned lc = 0, lr = 0;
            for (int t = 0; t < PER; ++t) {
                int j = tid + t * TT;
                if (j == idx) continue;
                if (!((act[j >> 5] >> (j & 31)) & 1u)) continue;
                float lx = fmaxf(Ax1, bxs[4 * j + 0]);
                float ly = fmaxf(Ay1, bxs[4 * j + 1]);
                float rx = fminf(Ax2, bxs[4 * j + 2]);
                float ry = fminf(Ay2, bxs[4 * j + 3]);
                float w  = fmaxf(rx - lx + 1.0f, 0.0f);
                float h  = fmaxf(ry - ly + 1.0f, 0.0f);
                float ov = w * h;
                float v  = ov / (Aar + area[j] - ov);
                if (v > NMS_THR) {
                    atomicAnd(&act[j >> 5], ~(1u << (j & 31)));  // suppress
                    ++lr;
                    int gj = agt[j];
                    if (gj != graw && v > giou[g * GG + max(gj, 0)]) {
                        ls += -logf(1.0f - v) * scr[j];
                        ++lc;
                    }
                }
            }
            // deterministic wave32 reduction, one slot per wave
            for (int off = 16; off; off >>= 1) {
                ls += __shfl_xor(ls, off, 32);
                lc += __shfl_xor(lc, off, 32);
                lr += __shfl_xor(lr, off, 32);
            }
            if ((tid & 31) == 0) { redS[wid] = ls; redC[wid] = lc; redR[wid] = lr; }
        }
        __syncthreads();
        if (tid == 0) {
            act[idx >> 5] &= ~(1u << (idx & 31));  // rem = active \ {idx}
            if (ai) {
                float    sum = 0.0f;
                unsigned cnt = 0, clr = 0;
                for (int w = 0; w < TT / 32; ++w) { sum += redS[w]; cnt += redC[w]; clr += redR[w]; }
                const int g    = graw;
                const int rec  = mrec[g];
                const int remc = M->activeCount - 1;
                if (rec >= 0) {
                    float ms = fmaxf(iou_box(bxs, area, rec, idx), EPSV);
                    if (remc > 0) M->tpull += -logf(1.0f - NMS_THR + ms) * scr[idx];
                    M->pc += 1.0f;
                } else {
                    mrec[g] = idx;
                }
                if (remc > 0 && cnt > 0) {
                    M->tpush += sum / (float)cnt;
                    M->pushc += (float)cnt;
                }
                M->activeCount = remc - (int)clr;
            }
        }
    }

    __syncthreads();
    if (tid == 0) {
        float pull = M->tpull / (M->pc + EPSV);
        float push = M->tpush / (M->pushc + EPSV);
        bool  ok   = (M->npos0 > 1);
        ws[2 * b + 0] = ok ? push : 0.0f;
        ws[2 * b + 1] = ok ? pull : 0.0f;
    }
}

__global__ void nms_finalize_kernel(const float* __restrict__ ws, float* __restrict__ out) {
    if (threadIdx.x == 0) {
        out[0] = 0.5f * (ws[0] + ws[2]);  // mean(push)
        out[1] = 0.5f * (ws[1] + ws[3]);  // mean(pull)
    }
}

extern "C" void kernel_launch(void* const* d_in, const int* in_sizes, int n_in,
                              void* d_out, int out_size, void* d_ws, size_t ws_size,
                              hipStream_t stream) {
    (void)in_sizes; (void)n_in; (void)out_size; (void)ws_size;
    // setup_inputs order: gt_inds (unused by the loss), anchor_gt_inds, gt_bboxes, proposals
    const int*   agt  = (const int*)d_in[1];
    const float* gtb  = (const float*)d_in[2];
    const float* prop = (const float*)d_in[3];
    float*       ws   = (float*)d_ws;
    float*       out  = (float*)d_out;

    nms_loss_kernel<<<dim3(BB), dim3(TT), SMEM_BYTES, stream>>>(agt, gtb, prop, ws);
    nms_finalize_kernel<<<dim3(1), dim3(32), 0, stream>>>(ws, out);
}